// GraphConvolution_214748364846
// MI455X (gfx1250) — compile-verified
//
#include <hip/hip_runtime.h>

typedef float v2f __attribute__((ext_vector_type(2)));
typedef float v8f __attribute__((ext_vector_type(8)));

#define C_DIM      128
#define KNE        16
#define TILE_N     16
#define LDS_STRIDE 132   // 128 + 4 floats padding -> conflict-free ds_load_b64

__global__ __launch_bounds__(256)
void graphconv_wmma_kernel(const float* __restrict__ input,
                           const int*   __restrict__ adj,
                           const float* __restrict__ w1,
                           const float* __restrict__ b1,
                           const float* __restrict__ w2,
                           const float* __restrict__ b2,
                           float* __restrict__ out,
                           int n_nodes)
{
    __shared__ float Xs[TILE_N * LDS_STRIDE];  // Xs[n*132 + c] : node's channel row
    __shared__ float As[TILE_N * LDS_STRIDE];  // neighbor-mean tile, same layout

    const int tid = threadIdx.x;          // 0..255 (8 waves of 32)
    const int n0  = blockIdx.x * TILE_N;

    // ---------- Phase 1a: stage X tile (128 ch x 16 nodes), transposed ----------
    for (int idx = tid; idx < C_DIM * TILE_N; idx += 256) {
        const int c  = idx >> 4;   // channel
        const int n  = idx & 15;   // node within tile
        const int gn = n0 + n;
        float v = 0.0f;
        if (gn < n_nodes) v = input[c * n_nodes + gn];
        Xs[n * LDS_STRIDE + c] = v;
    }

    // ---------- Phase 1b: neighbor gather + mean (L2-resident gather) ----------
    {
        const int n  = tid >> 4;   // node within tile   0..15
        const int cg = tid & 15;   // channel group       0..15 (8 channels each)
        const int gn = n0 + n;
        float acc[8];
        #pragma unroll
        for (int i = 0; i < 8; ++i) acc[i] = 0.0f;
        if (gn < n_nodes) {
            #pragma unroll 1
            for (int k = 0; k < KNE; ++k) {
                const int j = adj[gn * KNE + k];
                const float* col = input + j;
                #pragma unroll
                for (int i = 0; i < 8; ++i)
                    acc[i] += col[(cg * 8 + i) * n_nodes];
            }
        }
        #pragma unroll
        for (int i = 0; i < 8; ++i)
            As[n * LDS_STRIDE + cg * 8 + i] = acc[i] * (1.0f / KNE);
    }

    __syncthreads();

    // ---------- Phase 2: fused dual GEMM via V_WMMA_F32_16X16X4_F32 ----------
    // Wave w computes out-channel block [16w, 16w+16) x 16 nodes.
    // f32 16x4 A layout: lane L -> row M = L%16; VGPR j holds K = 2*(L>=16) + j.
    // B (4x16) mirrored: lane L -> col N = L%16; VGPR j holds K = 2*(L>=16) + j.
    const int lane  = tid & 31;
    const int wave  = tid >> 5;        // 0..7 -> M block
    const int m     = lane & 15;       // A row / B column / D node column
    const int khalf = lane >> 4;       // 0 or 1

    const float* w1row = w1 + (wave * 16 + m) * C_DIM + 2 * khalf;
    const float* w2row = w2 + (wave * 16 + m) * C_DIM + 2 * khalf;
    const float* xrow  = Xs + m * LDS_STRIDE + 2 * khalf;
    const float* arow  = As + m * LDS_STRIDE + 2 * khalf;

    v8f acc = {};
    #pragma unroll 4
    for (int kk = 0; kk < C_DIM; kk += 4) {
        v2f a1 = *(const v2f*)(w1row + kk);   // global_load_b64 (L2-hot weights)
        v2f bx = *(const v2f*)(xrow  + kk);   // ds_load_b64
        acc = __builtin_amdgcn_wmma_f32_16x16x4_f32(
                  false, a1, false, bx, (short)0, acc, false, false);
        v2f a2 = *(const v2f*)(w2row + kk);
        v2f ba = *(const v2f*)(arow  + kk);
        acc = __builtin_amdgcn_wmma_f32_16x16x4_f32(
                  false, a2, false, ba, (short)0, acc, false, false);
    }

    // ---------- Phase 3: bias + store ----------
    // D layout: acc[v] = D[M = wave*16 + 8*khalf + v][N = n0 + m]
    const int gn = n0 + m;
    if (gn < n_nodes) {
        #pragma unroll
        for (int v = 0; v < 8; ++v) {
            const int o = wave * 16 + khalf * 8 + v;
            out[o * n_nodes + gn] = acc[v] + b1[o] + b2[o];
        }
    }
}

extern "C" void kernel_launch(void* const* d_in, const int* in_sizes, int n_in,
                              void* d_out, int out_size, void* d_ws, size_t ws_size,
                              hipStream_t stream) {
    const float* input = (const float*)d_in[0];
    const int*   adj   = (const int*)  d_in[1];
    const float* w1    = (const float*)d_in[2];
    const float* b1    = (const float*)d_in[3];
    const float* w2    = (const float*)d_in[4];
    const float* b2    = (const float*)d_in[5];
    float*       out   = (float*)d_out;

    const int n_nodes = in_sizes[1] / KNE;   // adj has n_nodes * 16 entries
    const int blocks  = (n_nodes + TILE_N - 1) / TILE_N;

    graphconv_wmma_kernel<<<blocks, 256, 0, stream>>>(
        input, adj, w1, b1, w2, b2, out, n_nodes);
}